// NodeUpdateBlock_17815524344045
// MI455X (gfx1250) — compile-verified
//
#include <hip/hip_runtime.h>

// NodeUpdateBlock: 4 fused GEMMs (N x 640) @ (640 x 128) using V_WMMA_F32_16X16X4_F32.
// g=0: scalar channel (m0 @ W_lin0 + (f0 (x) attrs) @ W_tp0)
// g=1..3: vector component m=g-1 (m1[:,:,m] @ W_lin1 + (f1[:,:,m] (x) attrs) @ W_tp1)

#define ROWS_PER_BLOCK 64
#define RS 644              // padded LDS row stride (floats): conflict-free, 8B/16B aligned

typedef float v2f __attribute__((ext_vector_type(2)));
typedef float v8f __attribute__((ext_vector_type(8)));

__global__ __launch_bounds__(256)
void node_update_kernel(const float* __restrict__ m_i,
                        const float* __restrict__ node_feats,
                        const float* __restrict__ node_attrs,
                        const float* __restrict__ W_lin0,
                        const float* __restrict__ W_lin1,
                        const float* __restrict__ W_tp0,
                        const float* __restrict__ W_tp1,
                        float* __restrict__ out,
                        int N)
{
    extern __shared__ float smem[];
    float* act     = smem;                        // [64][RS] combined activations
    float* attrs_s = smem + ROWS_PER_BLOCK * RS;  // [64][4]

    const int b   = blockIdx.x;
    const int g   = b & 3;                  // 0 = scalar channel, 1..3 = vector comps
    const int n0  = (b >> 2) * ROWS_PER_BLOCK;
    const int tid = threadIdx.x;

    const float inv_lin = 0.08838834764831845f;   // 1/sqrt(128)
    const float inv_tp  = 0.04419417382415922f;   // 1/sqrt(512)

    // ---- stage node_attrs (64 rows x 4) ----
    {
        const int r = tid >> 2, v = tid & 3;
        const int n = n0 + r;
        attrs_s[tid] = (n < N) ? node_attrs[n * 4 + v] : 0.0f;
    }
    __syncthreads();

    // ---- fill act[:, 0:128]: scaled linear part ----
    #pragma unroll
    for (int j = 0; j < 32; ++j) {
        const int i = tid + j * 256;        // 64*128 elements
        const int r = i >> 7;
        const int k = i & 127;
        const int n = n0 + r;
        float v = 0.0f;
        if (n < N) {
            const int src = (g == 0) ? k : (128 + 3 * k + (g - 1));
            v = m_i[n * 512 + src] * inv_lin;
        }
        act[r * RS + k] = v;
    }

    // ---- fill act[:, 128:640]: scaled outer-product part f[u]*attrs[v] ----
    #pragma unroll
    for (int j = 0; j < 32; ++j) {
        const int i = tid + j * 256;        // 64*128 (r,u) tasks, 4 writes each
        const int r = i >> 7;
        const int u = i & 127;
        const int n = n0 + r;
        float f = 0.0f;
        if (n < N) {
            const int src = (g == 0) ? u : (128 + 3 * u + (g - 1));
            f = node_feats[n * 512 + src] * inv_tp;
        }
        float4 p;
        p.x = f * attrs_s[r * 4 + 0];
        p.y = f * attrs_s[r * 4 + 1];
        p.z = f * attrs_s[r * 4 + 2];
        p.w = f * attrs_s[r * 4 + 3];
        *(float4*)&act[r * RS + 128 + 4 * u] = p;   // 16B aligned: RS%4==0, 128+4u%4==0
    }
    __syncthreads();

    // ---- WMMA GEMM: 4 row-tiles x 1 col-tile per wave ----
    const int lane = tid & 31;
    const int wv   = tid >> 5;          // wave id = column tile (0..7)
    const int half = lane >> 4;         // 0: K{0,1}, 1: K{2,3}
    const int lcol = lane & 15;         // A row / B,D column within tile
    const int col  = wv * 16 + lcol;    // output column (0..127)
    const int kko  = half * 2;

    const float* __restrict__ Wlin = (g == 0) ? W_lin0 : W_lin1;
    const float* __restrict__ Wtp  = (g == 0) ? W_tp0  : W_tp1;

    v8f acc[4];
    #pragma unroll
    for (int rt = 0; rt < 4; ++rt) acc[rt] = (v8f){0,0,0,0,0,0,0,0};

    const float* aB = act + lcol * RS + kko;

    // K-loop over the linear part (K = 0..127), B rows from W_lin
    #pragma unroll 4
    for (int k = 0; k < 128; k += 4) {
        v2f bf;
        bf.x = Wlin[(k + kko)     * 128 + col];
        bf.y = Wlin[(k + kko + 1) * 128 + col];
        #pragma unroll
        for (int rt = 0; rt < 4; ++rt) {
            v2f a = *(const v2f*)(aB + rt * (16 * RS) + k);
            acc[rt] = __builtin_amdgcn_wmma_f32_16x16x4_f32(
                false, a, false, bf, (short)0, acc[rt], false, false);
        }
    }

    // K-loop over the tensor-product part (K = 128..639), B rows from W_tp
    const float* aB2 = aB + 128;
    #pragma unroll 4
    for (int k = 0; k < 512; k += 4) {
        v2f bf;
        bf.x = Wtp[(k + kko)     * 128 + col];
        bf.y = Wtp[(k + kko + 1) * 128 + col];
        #pragma unroll
        for (int rt = 0; rt < 4; ++rt) {
            v2f a = *(const v2f*)(aB2 + rt * (16 * RS) + k);
            acc[rt] = __builtin_amdgcn_wmma_f32_16x16x4_f32(
                false, a, false, bf, (short)0, acc[rt], false, false);
        }
    }

    // ---- scatter results: D VGPR i -> row (i + 8*half), col = lane&15 ----
    #pragma unroll
    for (int rt = 0; rt < 4; ++rt) {
        #pragma unroll
        for (int i = 0; i < 8; ++i) {
            const int n = n0 + rt * 16 + 8 * half + i;
            if (n < N) {
                const int idx = (g == 0) ? (n * 512 + col)
                                         : (n * 512 + 128 + col * 3 + (g - 1));
                out[idx] = acc[rt][i];
            }
        }
    }
}

extern "C" void kernel_launch(void* const* d_in, const int* in_sizes, int n_in,
                              void* d_out, int out_size, void* d_ws, size_t ws_size,
                              hipStream_t stream) {
    const float* m_i        = (const float*)d_in[0];
    const float* node_feats = (const float*)d_in[1];
    const float* node_attrs = (const float*)d_in[2];
    const float* W_lin0     = (const float*)d_in[3];
    const float* W_lin1     = (const float*)d_in[4];
    const float* W_tp0      = (const float*)d_in[5];
    const float* W_tp1      = (const float*)d_in[6];
    float* out = (float*)d_out;

    const int N = in_sizes[0] / 512;
    const int nRowBlocks = (N + ROWS_PER_BLOCK - 1) / ROWS_PER_BLOCK;
    const size_t ldsBytes = (size_t)(ROWS_PER_BLOCK * RS + 256) * sizeof(float);

    (void)hipFuncSetAttribute(reinterpret_cast<const void*>(node_update_kernel),
                              hipFuncAttributeMaxDynamicSharedMemorySize,
                              (int)ldsBytes);

    dim3 grid((unsigned)(nRowBlocks * 4));
    dim3 block(256);
    node_update_kernel<<<grid, block, ldsBytes, stream>>>(
        m_i, node_feats, node_attrs, W_lin0, W_lin1, W_tp0, W_tp1, out, N);
}